// VanillaVss_87771951661341
// MI455X (gfx1250) — compile-verified
//
#include <hip/hip_runtime.h>
#include <math.h>

// Problem constants (match reference)
#define CDIM   96
#define DLAB   32
#define NSTATE 16
#define DINX   192      // DIN
#define DTRX   6
#define DDIM   224      // D
#define HW     4096
#define L0X    64
#define LTOT   4160     // HW + L0
#define KK     4
#define NPROJ  38       // DTR + 2N
#define NPAD   48       // NPROJ padded to 3 tiles

typedef __attribute__((ext_vector_type(2))) float v2f;
typedef __attribute__((ext_vector_type(4))) float v4f;
typedef __attribute__((ext_vector_type(8))) float v8f;

// ---------------------------------------------------------------------------
// WMMA f32 16x16x4 tile helpers (CDNA5 VGPR layouts, ISA 7.12.2).
// All tiles are full 16x16; K dim is a compile-time constant -> full unroll,
// unconditional b64 loads, no EXEC juggling in the hot loop.
// ---------------------------------------------------------------------------

// A: (M x K) row-major, unit K stride.  B: (K x N) row-major, unit N stride.
template <int KDIM>
__device__ __forceinline__ v8f wmma_f32_tile_rowB(
    const float* __restrict__ A, long lda,
    const float* __restrict__ B, long ldb,
    int m0, int n0)
{
  const int lane = threadIdx.x & 31;
  const int half = lane >> 4;          // K-half: lanes 0-15 -> K=0,1 ; 16-31 -> K=2,3
  const int idx  = lane & 15;
  const float* ap = A + (long)(m0 + idx) * lda + 2 * half;
  const float* bp = B + (long)(2 * half) * ldb + (n0 + idx);
  v8f acc = {0.f, 0.f, 0.f, 0.f, 0.f, 0.f, 0.f, 0.f};
#pragma unroll
  for (int k0 = 0; k0 < KDIM; k0 += 4) {
    v2f a = *(const v2f*)(ap + k0);          // global_load_b64
    v2f b;
    b.x = bp[(long)(k0 + 0) * ldb];
    b.y = bp[(long)(k0 + 1) * ldb];
    acc = __builtin_amdgcn_wmma_f32_16x16x4_f32(
        false, a, false, b, (short)0, acc, false, false);
  }
  return acc;
}

// A: (M x K) row-major.  B given transposed: Bt is (N x K) row-major, so
// B[k][n] = Bt[n*ldbt + k]  -> both fragments are b64 loads.
template <int KDIM>
__device__ __forceinline__ v8f wmma_f32_tile_trB(
    const float* __restrict__ A, long lda,
    const float* __restrict__ Bt, long ldbt,
    int m0, int n0)
{
  const int lane = threadIdx.x & 31;
  const int half = lane >> 4;
  const int idx  = lane & 15;
  const float* ap = A + (long)(m0 + idx) * lda + 2 * half;
  const float* bp = Bt + (long)(n0 + idx) * ldbt + 2 * half;
  v8f acc = {0.f, 0.f, 0.f, 0.f, 0.f, 0.f, 0.f, 0.f};
#pragma unroll
  for (int k0 = 0; k0 < KDIM; k0 += 4) {
    v2f a = *(const v2f*)(ap + k0);
    v2f b = *(const v2f*)(bp + k0);
    acc = __builtin_amdgcn_wmma_f32_16x16x4_f32(
        false, a, false, b, (short)0, acc, false, false);
  }
  return acc;
}

// Full-tile store, row-major D with row stride ldd.
__device__ __forceinline__ void wmma_store_full(
    float* __restrict__ D, long ldd, int m0, int n0, v8f acc)
{
  const int lane = threadIdx.x & 31;
  const int half = lane >> 4;
  const int idx  = lane & 15;
  float* dp = D + (long)(m0 + 8 * half) * ldd + (n0 + idx);
#pragma unroll
  for (int r = 0; r < 8; ++r) dp[(long)r * ldd] = acc[r];
}

// Masked store (used only for the 38-row x_proj result).
__device__ __forceinline__ void wmma_store_mask(
    float* __restrict__ D, long ldd, int m0, int n0, v8f acc, int Mlim)
{
  const int lane = threadIdx.x & 31;
  const int half = lane >> 4;
  const int idx  = lane & 15;
#pragma unroll
  for (int r = 0; r < 8; ++r) {
    const int row = m0 + r + 8 * half;
    if (row < Mlim) D[(long)row * ldd + (n0 + idx)] = acc[r];
  }
}

// Transposed full store: D[col*cstride + row]; each lane owns 8 consecutive
// rows -> two 16-byte stores.
__device__ __forceinline__ void wmma_store_full_t(
    float* __restrict__ D, long cstride, int m0, int n0, v8f acc)
{
  const int lane = threadIdx.x & 31;
  const int half = lane >> 4;
  const int idx  = lane & 15;
  float* dp = D + (long)(n0 + idx) * cstride + (m0 + 8 * half);
  v4f lo = {acc[0], acc[1], acc[2], acc[3]};
  v4f hi = {acc[4], acc[5], acc[6], acc[7]};
  *(v4f*)(dp + 0) = lo;                      // global_store_b128
  *(v4f*)(dp + 4) = hi;
}

// ---------------------------------------------------------------------------
// 0) zero-pad x_proj_weight (K,38,224) -> (K,48,224) so GEMM tiles are full
// ---------------------------------------------------------------------------
__global__ void k_pad_xpw(const float* __restrict__ xpw, float* __restrict__ xpw_pad)
{
  const int idx = blockIdx.x * blockDim.x + threadIdx.x;
  if (idx >= KK * NPAD * DDIM) return;
  const int c = idx % DDIM;
  const int r = (idx / DDIM) % NPAD;
  const int k = idx / (NPAD * DDIM);
  xpw_pad[idx] = (r < NPROJ) ? xpw[((long)k * NPROJ + r) * DDIM + c] : 0.f;
}

// ---------------------------------------------------------------------------
// 1) xz = in_proj_w (384x96) @ feature (96x4096)
// ---------------------------------------------------------------------------
__global__ void k_in_proj(const float* __restrict__ W, const float* __restrict__ X,
                          float* __restrict__ Out)
{
  const int wid = blockIdx.x * (blockDim.x >> 5) + (threadIdx.x >> 5);
  const int mt = wid >> 8;            // 24 M-tiles
  const int nt = wid & 255;           // 256 N-tiles
  v8f acc = wmma_f32_tile_rowB<96>(W, 96, X, HW, mt * 16, nt * 16);
  wmma_store_full(Out, HW, mt * 16, nt * 16, acc);
}

// ---------------------------------------------------------------------------
// 2) depthwise 3x3 conv (SAME, zero pad) + bias + SiLU on xz[:192]
// ---------------------------------------------------------------------------
__global__ void k_conv_silu(const float* __restrict__ xz, const float* __restrict__ cw,
                            const float* __restrict__ cb, float* __restrict__ xact)
{
  const int idx = blockIdx.x * blockDim.x + threadIdx.x;
  if (idx >= DINX * HW) return;
  const int d = idx >> 12;
  const int p = idx & 4095;
  const int h = p >> 6, w = p & 63;
  float s = 0.f;
#pragma unroll
  for (int dy = 0; dy < 3; ++dy) {
    const int hy = h + dy - 1;
    if (hy < 0 || hy >= 64) continue;
#pragma unroll
    for (int dx = 0; dx < 3; ++dx) {
      const int wx = w + dx - 1;
      if (wx < 0 || wx >= 64) continue;
      s += xz[(long)d * HW + hy * 64 + wx] * cw[d * 9 + dy * 3 + dx];
    }
  }
  s += cb[d];
  xact[idx] = s / (1.f + __expf(-s));
}

// ---------------------------------------------------------------------------
// 3) build xs (K,D,L): 4 scan orderings + DLAB zero rows + label columns
// ---------------------------------------------------------------------------
__global__ void k_build_xs(const float* __restrict__ xact, const float* __restrict__ label,
                           float* __restrict__ xs)
{
  const long idx = (long)blockIdx.x * blockDim.x + threadIdx.x;
  if (idx >= (long)KK * DDIM * LTOT) return;
  const int l = (int)(idx % LTOT);
  const int t = (int)(idx / LTOT);
  const int d = t % DDIM;
  const int k = t / DDIM;
  float v = 0.f;
  if (l < HW) {
    if (d < DINX) {
      int p;
      if (k == 0) p = l;
      else if (k == 1) { const int w = l >> 6, h = l & 63; p = h * 64 + w; }
      else if (k == 2) p = HW - 1 - l;
      else { const int lr = HW - 1 - l; const int w = lr >> 6, h = lr & 63; p = h * 64 + w; }
      v = xact[(long)d * HW + p];
    }
  } else {
    const int j = l - HW;
    if (d >= DINX) v = label[j * DLAB + (d - DINX)];
  }
  xs[idx] = v;
}

// ---------------------------------------------------------------------------
// 4) x_dbl[k] = xpw_pad[k] (48x224) @ xs[k] (224x4160); store rows [0,38)
// ---------------------------------------------------------------------------
__global__ void k_xdbl(const float* __restrict__ xpw_pad, const float* __restrict__ xs,
                       float* __restrict__ xdbl)
{
  const int wid = blockIdx.x * (blockDim.x >> 5) + (threadIdx.x >> 5);
  const int k   = wid / 780;           // 3 Mtiles * 260 Ntiles per k
  const int rem = wid % 780;
  const int mt  = rem / 260;
  const int nt  = rem % 260;
  const float* A = xpw_pad + (long)k * NPAD * DDIM;
  const float* B = xs + (long)k * DDIM * LTOT;
  float* Dst = xdbl + (long)k * NPROJ * LTOT;
  v8f acc = wmma_f32_tile_rowB<DDIM>(A, DDIM, B, LTOT, mt * 16, nt * 16);
  wmma_store_mask(Dst, LTOT, mt * 16, nt * 16, acc, NPROJ);
}

// ---------------------------------------------------------------------------
// 5) delta = softplus(dt_projs_weight[k] @ x_dbl[k][:6] + bias)
// ---------------------------------------------------------------------------
__global__ void k_delta(const float* __restrict__ xdbl, const float* __restrict__ dtw,
                        const float* __restrict__ dtb, float* __restrict__ delta)
{
  const long idx = (long)blockIdx.x * blockDim.x + threadIdx.x;
  if (idx >= (long)KK * DDIM * LTOT) return;
  const int l = (int)(idx % LTOT);
  const int t = (int)(idx / LTOT);
  const int d = t % DDIM;
  const int k = t / DDIM;
  const float* xb = xdbl + (long)k * NPROJ * LTOT + l;
  const float* wp = dtw + ((long)k * DDIM + d) * DTRX;
  float s = 0.f;
#pragma unroll
  for (int r = 0; r < DTRX; ++r) s += xb[(long)r * LTOT] * wp[r];
  s += dtb[k * DDIM + d];
  delta[idx] = (s > 20.f) ? s : log1pf(__expf(s));
}

// ---------------------------------------------------------------------------
// 6) selective scan: wave = 2 channels x 16 state-lanes, sequential over L
// ---------------------------------------------------------------------------
__global__ void k_scan(const float* __restrict__ xs, const float* __restrict__ xdbl,
                       const float* __restrict__ delta, const float* __restrict__ A_logs,
                       const float* __restrict__ Ds, float* __restrict__ y)
{
  const int wid  = blockIdx.x * (blockDim.x >> 5) + (threadIdx.x >> 5);
  const int lane = threadIdx.x & 31;
  const int sub  = lane >> 4;          // which channel of the pair
  const int n    = lane & 15;          // state index
  const int ch   = wid * 2 + sub;      // 0..895 = k*DDIM + d
  if (ch >= KK * DDIM) return;
  const int k = ch / DDIM;

  const float Av = -__expf(A_logs[(long)ch * NSTATE + n]);
  const float Dv = Ds[ch];
  const float* dp = delta + (long)ch * LTOT;
  const float* xp = xs + (long)ch * LTOT;
  const float* bp = xdbl + (long)k * NPROJ * LTOT + (long)(DTRX + n) * LTOT;
  const float* cp = xdbl + (long)k * NPROJ * LTOT + (long)(DTRX + NSTATE + n) * LTOT;
  float* yo = y + (long)ch * LTOT;

  float h = 0.f;
  for (int l = 0; l < LTOT; ++l) {
    const float dv = dp[l];
    const float xv = xp[l];
    const float dA = __expf(dv * Av);
    h = dA * h + dv * xv * bp[l];
    float t = h * cp[l];
    t += __shfl_xor(t, 8, 16);
    t += __shfl_xor(t, 4, 16);
    t += __shfl_xor(t, 2, 16);
    t += __shfl_xor(t, 1, 16);
    if (n == 0) yo[l] = t + Dv * xv;
  }
}

// ---------------------------------------------------------------------------
// 7) undo the 4 scan orderings, sum, transpose -> ysum (HW x D)
// ---------------------------------------------------------------------------
__global__ void k_combine(const float* __restrict__ y, float* __restrict__ ysum)
{
  const int idx = blockIdx.x * blockDim.x + threadIdx.x;
  if (idx >= HW * DDIM) return;
  const int d = idx % DDIM;
  const int p = idx / DDIM;
  const int h = p >> 6, w = p & 63;
  const long base = (long)d * LTOT;
  const long ks = (long)DDIM * LTOT;
  const float t = y[0 * ks + base + p]
                + y[1 * ks + base + (w * 64 + h)]
                + y[2 * ks + base + (HW - 1 - p)]
                + y[3 * ks + base + (HW - 1 - (w * 64 + h))];
  ysum[(long)p * DDIM + d] = t;
}

// ---------------------------------------------------------------------------
// 8) yp = ysum (4096x224) @ pre_scale_w.T ; pre_scale_w is (192x224) = B^T
// ---------------------------------------------------------------------------
__global__ void k_prescale(const float* __restrict__ ysum, const float* __restrict__ psw,
                           float* __restrict__ yp)
{
  const int wid = blockIdx.x * (blockDim.x >> 5) + (threadIdx.x >> 5);
  const int mt = wid / 12;             // 256 M-tiles
  const int nt = wid % 12;             // 12 N-tiles
  v8f acc = wmma_f32_tile_trB<DDIM>(ysum, DDIM, psw, DDIM, mt * 16, nt * 16);
  wmma_store_full(yp, DINX, mt * 16, nt * 16, acc);
}

// ---------------------------------------------------------------------------
// 9) LayerNorm over DIN per pixel, then gate with silu(z). One wave per pixel.
// ---------------------------------------------------------------------------
__global__ void k_ln_gate(const float* __restrict__ ypbuf, const float* __restrict__ xz,
                          const float* __restrict__ lnw, const float* __restrict__ lnb,
                          float* __restrict__ yng)
{
  const int p = blockIdx.x * (blockDim.x >> 5) + (threadIdx.x >> 5);
  const int lane = threadIdx.x & 31;
  if (p >= HW) return;
  const float* row = ypbuf + (long)p * DINX;
  float s = 0.f, ss = 0.f;
#pragma unroll
  for (int c0 = 0; c0 < DINX; c0 += 32) {
    const float v = row[c0 + lane];
    s += v; ss += v * v;
  }
#pragma unroll
  for (int off = 16; off > 0; off >>= 1) {
    s  += __shfl_xor(s,  off, 32);
    ss += __shfl_xor(ss, off, 32);
  }
  const float mu  = s * (1.0f / DINX);
  const float var = ss * (1.0f / DINX) - mu * mu;
  const float inv = rsqrtf(var + 1e-5f);
#pragma unroll
  for (int c0 = 0; c0 < DINX; c0 += 32) {
    const int c = c0 + lane;
    const float v = (row[c] - mu) * inv * lnw[c] + lnb[c];
    const float z = xz[(long)(DINX + c) * HW + p];
    const float sz = z / (1.f + __expf(-z));
    yng[(long)p * DINX + c] = v * sz;
  }
}

// ---------------------------------------------------------------------------
// 10) out[c][p] = (yng (4096x192) @ out_proj_w.T) ; out_proj_w (96x192) = B^T
//     transposed b128 store into d_out (C,H,W)
// ---------------------------------------------------------------------------
__global__ void k_outproj(const float* __restrict__ yng, const float* __restrict__ opw,
                          float* __restrict__ out)
{
  const int wid = blockIdx.x * (blockDim.x >> 5) + (threadIdx.x >> 5);
  const int mt = wid / 6;              // 256 M-tiles (pixels)
  const int nt = wid % 6;              // 6 N-tiles (channels)
  v8f acc = wmma_f32_tile_trB<DINX>(yng, DINX, opw, DINX, mt * 16, nt * 16);
  wmma_store_full_t(out, HW, mt * 16, nt * 16, acc);
}

// ---------------------------------------------------------------------------
extern "C" void kernel_launch(void* const* d_in, const int* in_sizes, int n_in,
                              void* d_out, int out_size, void* d_ws, size_t ws_size,
                              hipStream_t stream)
{
  const float* feature         = (const float*)d_in[0];
  const float* label           = (const float*)d_in[1];
  const float* in_proj_w       = (const float*)d_in[2];
  const float* conv_w          = (const float*)d_in[3];
  const float* conv_b          = (const float*)d_in[4];
  const float* x_proj_weight   = (const float*)d_in[5];
  const float* dt_projs_weight = (const float*)d_in[6];
  const float* dt_projs_bias   = (const float*)d_in[7];
  const float* A_logs          = (const float*)d_in[8];
  const float* Ds              = (const float*)d_in[9];
  const float* pre_scale_w     = (const float*)d_in[10];
  const float* ln_w            = (const float*)d_in[11];
  const float* ln_b            = (const float*)d_in[12];
  const float* out_proj_w      = (const float*)d_in[13];
  float* out = (float*)d_out;
  float* ws  = (float*)d_ws;

  // workspace partition (floats); total ~16.7M floats (~67 MB), all even
  // offsets so v2f/v4f accesses stay >=8B aligned.
  float* xz      = ws;                                 // 384*4096
  float* xact    = xz      + (long)384 * HW;           // 192*4096
  float* xs      = xact    + (long)DINX * HW;          // 4*224*4160
  float* xdbl    = xs      + (long)KK * DDIM * LTOT;   // 4*38*4160
  float* delta   = xdbl    + (long)KK * NPROJ * LTOT;  // 4*224*4160
  float* ybuf    = delta   + (long)KK * DDIM * LTOT;   // 4*224*4160
  float* ysum    = ybuf    + (long)KK * DDIM * LTOT;   // 4096*224
  float* ypbuf   = ysum    + (long)HW * DDIM;          // 4096*192
  float* yng     = ypbuf   + (long)HW * DINX;          // 4096*192
  float* xpw_pad = yng     + (long)HW * DINX;          // 4*48*224

  (void)in_sizes; (void)n_in; (void)out_size; (void)ws_size;

  // 0) pad x_proj weights to full tiles
  k_pad_xpw<<<(KK * NPAD * DDIM + 255) / 256, 256, 0, stream>>>(x_proj_weight, xpw_pad);
  // 1) in_proj GEMM: 24x256 tiles, 4 waves/block
  k_in_proj<<<1536, 128, 0, stream>>>(in_proj_w, feature, xz);
  // 2) depthwise conv + SiLU
  k_conv_silu<<<(DINX * HW + 255) / 256, 256, 0, stream>>>(xz, conv_w, conv_b, xact);
  // 3) build xs
  {
    long n = (long)KK * DDIM * LTOT;
    k_build_xs<<<(int)((n + 255) / 256), 256, 0, stream>>>(xact, label, xs);
  }
  // 4) x_proj GEMMs: 4 * 3 * 260 tiles
  k_xdbl<<<780, 128, 0, stream>>>(xpw_pad, xs, xdbl);
  // 5) delta = softplus(dt-proj + bias)
  {
    long n = (long)KK * DDIM * LTOT;
    k_delta<<<(int)((n + 255) / 256), 256, 0, stream>>>(xdbl, dt_projs_weight, dt_projs_bias, delta);
  }
  // 6) selective scan: 448 waves (896 channels), 4 waves/block
  k_scan<<<112, 128, 0, stream>>>(xs, xdbl, delta, A_logs, Ds, ybuf);
  // 7) combine 4 directions + transpose
  k_combine<<<(HW * DDIM + 255) / 256, 256, 0, stream>>>(ybuf, ysum);
  // 8) pre_scale GEMM: 256x12 tiles
  k_prescale<<<768, 128, 0, stream>>>(ysum, pre_scale_w, ypbuf);
  // 9) LayerNorm + silu gate: 1 wave / pixel
  k_ln_gate<<<512, 256, 0, stream>>>(ypbuf, xz, ln_w, ln_b, yng);
  // 10) out_proj GEMM with transposed b128 store: 256x6 tiles
  k_outproj<<<384, 128, 0, stream>>>(yng, out_proj_w, out);
}